// ROIPooling_18571438588668
// MI455X (gfx1250) — compile-verified
//
#include <hip/hip_runtime.h>
#include <hip/hip_bf16.h>
#include <stdint.h>

// Problem constants (from reference):
#define B_    8
#define H_    160
#define W_    160
#define C_    256
#define NROI  1024
#define CROPD 14

// Byte strides of NHWC feature map
#define IMG_STRIDE  ((unsigned long long)H_ * W_ * C_ * 4ull)   // per batch image
#define ROW_STRIDE  ((unsigned long long)W_ * C_ * 4ull)        // per y row

typedef unsigned int uint32x4_t_ __attribute__((ext_vector_type(4)));
typedef int          int32x8_t_  __attribute__((ext_vector_type(8)));
typedef int          int32x4_t_  __attribute__((ext_vector_type(4)));

__global__ __launch_bounds__(256)
void roi_pool_tdm_kernel(const float* __restrict__ inputs,
                         const float* __restrict__ rois,
                         const float* __restrict__ im_info,
                         float* __restrict__ out)
{
    // LDS: 8 gather slots = crop-row{h0,h1} x src-row{yl,yh} x half{ix0-6,ix7-13}
    // each slot: 14 gathered channel-vectors x 256 ch x 4B = 14 KB. Total 112 KB.
    __shared__ float sm[8 * 14 * C_];

    const int py  = blockIdx.x;    // pooled row 0..6
    const int roi = blockIdx.y;    // 0..1023
    const int c   = threadIdx.x;   // channel 0..255

    // ROI box -> normalized (y1,x1,y2,x2) exactly like the reference
    const float r0   = rois[roi * 5 + 0];
    const float rx1  = rois[roi * 5 + 1];
    const float ry1  = rois[roi * 5 + 2];
    const float rx2  = rois[roi * 5 + 3];
    const float ry2  = rois[roi * 5 + 4];
    const float Himg = im_info[roi * 2 + 0];
    const float Wimg = im_info[roi * 2 + 1];
    const int   bimg = (int)r0;

    const float y1n = ry1 / Himg, y2n = ry2 / Himg;
    const float x1n = rx1 / Wimg, x2n = rx2 / Wimg;
    const float base_y  = y1n * (float)(H_ - 1);
    const float scale_y = (y2n - y1n) * (float)(H_ - 1) / (float)(CROPD - 1);
    const float base_x  = x1n * (float)(W_ - 1);
    const float scale_x = (x2n - x1n) * (float)(W_ - 1) / (float)(CROPD - 1);

    // Per-crop-row (h = 0,1) vertical params
    float ylerp[2];
    bool  vy[2];
    int   yl[2], yh[2];
#pragma unroll
    for (int h = 0; h < 2; ++h) {
        const int   iy = 2 * py + h;
        const float y  = fmaf((float)iy, scale_y, base_y);
        const float yf = floorf(y);
        ylerp[h] = y - yf;
        vy[h]    = (y >= 0.0f) && (y <= (float)(H_ - 1));
        yl[h]    = (int)fminf(fmaxf(yf,        0.0f), (float)(H_ - 1));
        yh[h]    = (int)fminf(fmaxf(yf + 1.0f, 0.0f), (float)(H_ - 1));
    }

    float acc[7];
#pragma unroll
    for (int i = 0; i < 7; ++i) acc[i] = -__builtin_huge_valf();

    // LDS byte offset of sm[] (addrspace(3) pointer value == LDS byte offset)
    const unsigned lds_base =
        (unsigned)(unsigned long long)(__attribute__((address_space(3))) char*)(&sm[0]);

    // ---- Issue ALL 8 gather descriptors up front (wave 0 drives the TDM).
    // Tensor ops from one wave complete in order, so s_wait_tensorcnt(4)
    // releases crop-row 0's slots while crop-row 1 is still streaming.
    if (threadIdx.x < 32) {
#pragma unroll
        for (int h = 0; h < 2; ++h) {
            const int rows01[2] = { yl[h], yh[h] };
#pragma unroll
            for (int r01 = 0; r01 < 2; ++r01) {
                const unsigned long long gaddr =
                    (unsigned long long)(uintptr_t)inputs +
                    (unsigned long long)(unsigned)bimg        * IMG_STRIDE +
                    (unsigned long long)(unsigned)rows01[r01] * ROW_STRIDE;
#pragma unroll
                for (int half = 0; half < 2; ++half) {
                    // 14 gather row-indices = (xl,xh) pairs for ix = 7*half .. 7*half+6
                    unsigned idx_dw[8];
#pragma unroll
                    for (int k = 0; k < 7; ++k) {
                        const int   ix = half * 7 + k;
                        const float x  = fmaf((float)ix, scale_x, base_x);
                        const float xf = floorf(x);
                        const int xlo = (int)fminf(fmaxf(xf,        0.0f), (float)(W_ - 1));
                        const int xhi = (int)fminf(fmaxf(xf + 1.0f, 0.0f), (float)(W_ - 1));
                        idx_dw[k] = (unsigned)xlo | ((unsigned)xhi << 16);
                    }
                    idx_dw[7] = 0u;

                    const int slot = h * 4 + r01 * 2 + half;   // 0..7
                    // ---- D# group 0: gather_mode=1, 16-bit idx, count=1 ----
                    uint32x4_t_ g0;
                    g0[0] = 0x80000001u;
                    g0[1] = (unsigned)__builtin_amdgcn_readfirstlane(
                                (int)(lds_base + (unsigned)slot * (14u * C_ * 4u)));
                    g0[2] = (unsigned)__builtin_amdgcn_readfirstlane((int)(unsigned)gaddr);
                    g0[3] = (unsigned)__builtin_amdgcn_readfirstlane(
                                (int)((((unsigned)(gaddr >> 32)) & 0x01FFFFFFu) | (2u << 30)));
                    // ---- D# group 1 ----
                    int32x8_t_ g1;
                    g1[0] = 0x00020000;        // workgroup_mask=0, data_size=4B
                    g1[1] = (int)(256u << 16); // tensor_dim0 = 256 elems (bits 79:48 lo)
                    g1[2] = (int)(160u << 16); // tensor_dim0 hi=0 | tensor_dim1 = 160 rows
                    g1[3] = (int)(256u << 16); // tensor_dim1 hi=0 | tile_dim0 = 256
                    g1[4] = 14;                // tile_dim1 = #valid gather indices
                    g1[5] = 256;               // tensor_dim0_stride = 256 elems (1KB rows)
                    g1[6] = 0;
                    g1[7] = 0;
                    // ---- D# groups 2/3: packed 16-bit row indices ----
                    int32x4_t_ g2, g3;
#pragma unroll
                    for (int k = 0; k < 4; ++k)
                        g2[k] = __builtin_amdgcn_readfirstlane((int)idx_dw[k]);
#pragma unroll
                    for (int k = 0; k < 4; ++k)
                        g3[k] = __builtin_amdgcn_readfirstlane((int)idx_dw[4 + k]);

                    // clang-23 / therock-10.0 builtin: 6 args; 5th (int32x8)
                    // is beyond the ISA's 4 descriptor groups -> zero-filled.
                    int32x8_t_ gX;
#pragma unroll
                    for (int k = 0; k < 8; ++k) gX[k] = 0;
                    __builtin_amdgcn_tensor_load_to_lds(g0, g1, g2, g3, gX, 0);
                }
            }
        }
    }

    // ---- Pipelined consume: wait for first 4 tensor ops, compute crop row 0
    // while crop row 1 is still in flight, then drain and compute row 1.
#pragma unroll
    for (int h = 0; h < 2; ++h) {
        if (threadIdx.x < 32) {
            if (h == 0) __builtin_amdgcn_s_wait_tensorcnt(4);
            else        __builtin_amdgcn_s_wait_tensorcnt(0);
        }
        __syncthreads();

        const int sb = h * 4;     // slot base for this crop row
#pragma unroll
        for (int ix = 0; ix < CROPD; ++ix) {
            const float x     = fmaf((float)ix, scale_x, base_x);
            const float xf    = floorf(x);
            const float xlerp = x - xf;
            const bool  valid = vy[h] && (x >= 0.0f) && (x <= (float)(W_ - 1));
            const int   half  = (ix >= 7) ? 1 : 0;
            const int   j     = ix - 7 * half;

            const float tl = sm[((sb + 0 + half) * 14 + 2 * j + 0) * C_ + c];
            const float tr = sm[((sb + 0 + half) * 14 + 2 * j + 1) * C_ + c];
            const float bl = sm[((sb + 2 + half) * 14 + 2 * j + 0) * C_ + c];
            const float br = sm[((sb + 2 + half) * 14 + 2 * j + 1) * C_ + c];

            const float top = fmaf(tr - tl, xlerp, tl);
            const float bot = fmaf(br - bl, xlerp, bl);
            float v = fmaf(bot - top, ylerp[h], top);
            v = valid ? v : 0.0f;
            acc[ix >> 1] = fmaxf(acc[ix >> 1], v);
        }
    }

    // Coalesced stores: out[roi, py, px, c]
#pragma unroll
    for (int px = 0; px < 7; ++px) {
        out[(((roi * 7 + py) * 7) + px) * C_ + c] = acc[px];
    }
}

extern "C" void kernel_launch(void* const* d_in, const int* in_sizes, int n_in,
                              void* d_out, int out_size, void* d_ws, size_t ws_size,
                              hipStream_t stream)
{
    (void)in_sizes; (void)n_in; (void)out_size; (void)d_ws; (void)ws_size;
    const float* inputs  = (const float*)d_in[0];
    const float* rois    = (const float*)d_in[1];
    const float* im_info = (const float*)d_in[2];
    float* out = (float*)d_out;

    dim3 grid(7, NROI, 1);
    dim3 block(256, 1, 1);
    hipLaunchKernelGGL(roi_pool_tdm_kernel, grid, block, 0, stream,
                       inputs, rois, im_info, out);
}